// GNNFeatureExtractor_57930518888449
// MI455X (gfx1250) — compile-verified
//
#include <hip/hip_runtime.h>
#include <hip/hip_bf16.h>
#include <cstdint>

typedef __attribute__((ext_vector_type(2))) float v2f;
typedef __attribute__((ext_vector_type(8))) float v8f;

// ---------------------------------------------------------------------------
// Degree / normalization kernels
// ---------------------------------------------------------------------------
__global__ void init_deg_kernel(float* __restrict__ deg, int n) {
    int i = blockIdx.x * blockDim.x + threadIdx.x;
    if (i < n) deg[i] = 1.0f;  // self-loop contributes 1
}

__global__ void count_deg_kernel(const long long* __restrict__ dst,
                                 float* __restrict__ deg, long long nE) {
    long long e = (long long)blockIdx.x * blockDim.x + threadIdx.x;
    if (e < nE) atomicAdd(&deg[(int)dst[e]], 1.0f);
}

__global__ void make_dinv_kernel(float* __restrict__ deg, int n) {
    int i = blockIdx.x * blockDim.x + threadIdx.x;
    if (i < n) deg[i] = rsqrtf(deg[i]);  // deg >= 1 always (self-loop)
}

// ---------------------------------------------------------------------------
// WMMA f32 GEMM with fused row scaling and fused self-loop init:
//   val = dinv[r] * (X @ W)[r,:] ;  hs = val ;  agg = val   (agg starts as I*hs)
// X: [nRows, K], W: [K, 64]. One wave -> one 16x16 output tile.
// Uses V_WMMA_F32_16X16X4_F32 (exact fp32, matches reference numerics).
// ---------------------------------------------------------------------------
template <int K>
__global__ __launch_bounds__(256) void gcn_gemm_rowscale(
    const float* __restrict__ X, const float* __restrict__ W,
    const float* __restrict__ dinv, float* __restrict__ hs,
    float* __restrict__ aggv, int nRows, int nTilesM) {
    const int wave = (int)((blockIdx.x * blockDim.x + threadIdx.x) >> 5);
    const int lane = (int)(threadIdx.x & 31);
    const int tileM = wave >> 2;       // 64 cols / 16 = 4 col-tiles per row-tile
    if (tileM >= nTilesM) return;      // wave-uniform: EXEC stays all-ones
    const int col0 = (wave & 3) * 16;
    const int row0 = tileM * 16;
    const int mn = lane & 15;          // M (for A) / N (for B,C,D)
    const int half = lane >> 4;        // lane half selects K sub-pair
    const int khalf = half * 2;        // K offset {0,2} within each 4-K block

    // A fragment: lane holds X[row0+mn][kk+khalf .. kk+khalf+1]
    const int rIn = min(row0 + mn, nRows - 1);
    const float* xrow = X + (size_t)rIn * K + khalf;
    // B fragment: lane holds W[kk+khalf .. kk+khalf+1][col0+mn]
    const float* wcol = W + col0 + mn + (size_t)khalf * 64;

    v8f c = {};
#pragma unroll
    for (int kk = 0; kk < K; kk += 4) {
        v2f a, b;
        a.x = xrow[kk];
        a.y = xrow[kk + 1];
        b.x = wcol[(size_t)kk * 64];
        b.y = wcol[(size_t)kk * 64 + 64];
        c = __builtin_amdgcn_wmma_f32_16x16x4_f32(
            /*neg_a=*/false, a, /*neg_b=*/false, b,
            /*c_mod=*/(short)0, c, /*reuse_a=*/false, /*reuse_b=*/false);
    }

    // C/D layout: VGPR v -> M = v + 8*half, N = mn
    const int rbase = row0 + half * 8;
    if (row0 + 16 <= nRows) {
        // Full tile: wave-uniform fast path, no per-element EXEC churn.
#pragma unroll
        for (int v = 0; v < 8; ++v) {
            const int r = rbase + v;
            const float val = c[v] * dinv[r];
            const size_t o = (size_t)r * 64 + col0 + mn;
            hs[o] = val;
            aggv[o] = val;
        }
    } else {
        // Ragged tail tile (not taken when nRows % 16 == 0).
#pragma unroll
        for (int v = 0; v < 8; ++v) {
            const int r = rbase + v;
            if (r < nRows) {
                const float val = c[v] * dinv[r];
                const size_t o = (size_t)r * 64 + col0 + mn;
                hs[o] = val;
                aggv[o] = val;
            }
        }
    }
}

// ---------------------------------------------------------------------------
// Edge scatter: agg[dst, f] += hs[src, f]; one thread per (edge, feature).
// Feature arrays (25.6 MB) are L2-resident (192 MB L2) -> L2 atomic units.
// ---------------------------------------------------------------------------
__global__ void scatter_kernel(const long long* __restrict__ src,
                               const long long* __restrict__ dst,
                               const float* __restrict__ hs,
                               float* __restrict__ aggv, long long nE) {
    long long t = (long long)blockIdx.x * blockDim.x + threadIdx.x;
    long long e = t >> 6;
    if (e >= nE) return;
    int f = (int)(t & 63);
    int s = (int)src[e];
    int d = (int)dst[e];
    atomicAdd(&aggv[(size_t)d * 64 + f], hs[(size_t)s * 64 + f]);
}

// ---------------------------------------------------------------------------
// out = relu(dinv ⊙ agg + b)
// ---------------------------------------------------------------------------
__global__ void finish_kernel(const float* __restrict__ aggv,
                              const float* __restrict__ dinv,
                              const float* __restrict__ bias,
                              float* __restrict__ out, long long total) {
    long long i = (long long)blockIdx.x * blockDim.x + threadIdx.x;
    if (i >= total) return;
    int f = (int)(i & 63);
    long long node = i >> 6;
    float v = dinv[node] * aggv[i] + bias[f];
    out[i] = fmaxf(v, 0.0f);
}

// ---------------------------------------------------------------------------
extern "C" void kernel_launch(void* const* d_in, const int* in_sizes, int n_in,
                              void* d_out, int out_size, void* d_ws,
                              size_t ws_size, hipStream_t stream) {
    const float* x = (const float*)d_in[0];
    const long long* ei = (const long long*)d_in[1];  // int64 [2, E]
    const float* W1 = (const float*)d_in[2];
    const float* b1 = (const float*)d_in[3];
    const float* W2 = (const float*)d_in[4];
    const float* b2 = (const float*)d_in[5];

    const int N = in_sizes[0] / 32;            // 100000
    const long long E = in_sizes[1] / 2;       // 1600000
    const long long NF = (long long)N * 64;

    const long long* srcIdx = ei;
    const long long* dstIdx = ei + E;

    // Workspace layout: dinv[N] | hs[N*64] | agg[N*64]  (~51.6 MB)
    float* dinv = (float*)d_ws;
    float* hs = dinv + N;
    float* aggv = hs + NF;

    float* outp = (float*)d_out;

    const int B = 256;
    const int nTilesM = (N + 15) / 16;
    const int gemmBlocks = (nTilesM * 4 + 7) / 8;  // 8 waves / 256-thread block
    const int nBlocksN = (N + B - 1) / B;
    const long long scatterThreads = E * 64;
    const int scatterBlocks = (int)((scatterThreads + B - 1) / B);
    const int edgeBlocks = (int)((E + B - 1) / B);
    const int nfBlocks = (int)((NF + B - 1) / B);

    // --- normalization: deg -> dinv (in place) ---
    init_deg_kernel<<<nBlocksN, B, 0, stream>>>(dinv, N);
    count_deg_kernel<<<edgeBlocks, B, 0, stream>>>(dstIdx, dinv, E);
    make_dinv_kernel<<<nBlocksN, B, 0, stream>>>(dinv, N);

    // --- layer 1: hs = agg = dinv ⊙ (x @ W1); agg += A·hs; h1 = relu(dinv⊙agg+b1)
    gcn_gemm_rowscale<32><<<gemmBlocks, B, 0, stream>>>(x, W1, dinv, hs, aggv,
                                                        N, nTilesM);
    scatter_kernel<<<scatterBlocks, B, 0, stream>>>(srcIdx, dstIdx, hs, aggv, E);
    finish_kernel<<<nfBlocks, B, 0, stream>>>(aggv, dinv, b1, outp, NF);

    // --- layer 2: input = d_out (h1), output overwrites d_out ---
    gcn_gemm_rowscale<64><<<gemmBlocks, B, 0, stream>>>(outp, W2, dinv, hs, aggv,
                                                        N, nTilesM);
    scatter_kernel<<<scatterBlocks, B, 0, stream>>>(srcIdx, dstIdx, hs, aggv, E);
    finish_kernel<<<nfBlocks, B, 0, stream>>>(aggv, dinv, b2, outp, NF);
}